// scGraph_39986145526072
// MI455X (gfx1250) — compile-verified
//
#include <hip/hip_runtime.h>
#include <hip/hip_bf16.h>
#include <math.h>

typedef float v2f __attribute__((ext_vector_type(2)));
typedef float v8f __attribute__((ext_vector_type(8)));
typedef int b128_t __attribute__((__vector_size__(4 * sizeof(int))));

#define AS1 __attribute__((address_space(1)))
#define AS3 __attribute__((address_space(3)))

#if defined(__has_builtin)
#if __has_builtin(__builtin_amdgcn_global_load_async_to_lds_b128)
#define HAS_ASYNC_LDS 1
#endif
#endif

#define NN   3000
#define EE   96000
#define BB   64
#define CIN  8
#define MIDC 16
#define G1C  32
#define G2C  8
#define FC1C 512
#define FC2C 128
#define OUTC 10
#define D0   24000   // G2C * NN
#define EPSV 1e-5f

#define KC       64            // K-chunk staged in LDS per buffer
#define LDSTRIDE 68            // KC + 4 floats pad: 16B-aligned rows, conflict-free b64 reads

__device__ __forceinline__ float sigmoidf_(float x) { return 1.0f / (1.0f + __expf(-x)); }

// ---------------- zero init for atomic accumulators ----------------
__global__ void k_zero(float* __restrict__ p, int n) {
    int i = blockIdx.x * blockDim.x + threadIdx.x;
    if (i < n) p[i] = 0.0f;
}

// ---------------- edge gather + weighted scatter-sum (L2-resident atomics) ----------------
__global__ void k_scatter(const float* __restrict__ x, const int* __restrict__ ei,
                          const float* __restrict__ ewr,
                          float* __restrict__ summed, float* __restrict__ cnt) {
    long long tid = (long long)blockIdx.x * blockDim.x + threadIdx.x;
    const long long total = (long long)EE * 128;
    if (tid >= total) return;
    int e = (int)(tid >> 7);
    int i = (int)(tid & 127);
    int row = ei[e];
    int col = ei[EE + e];
    float w = sigmoidf_(ewr[e]);
    const float4 v = *(const float4*)(x + (size_t)row * 512 + i * 4);
    float* dst = summed + (size_t)col * 512 + i * 4;
    atomicAdd(dst + 0, v.x * w);
    atomicAdd(dst + 1, v.y * w);
    atomicAdd(dst + 2, v.z * w);
    atomicAdd(dst + 3, v.w * w);
    if (i == 0) atomicAdd(cnt + col, 1.0f);
}

// ---------------- agg-mean -> sage matmul -> relu ; accumulate LN stats ----------------
__global__ void k_sage(const float* __restrict__ summed, const float* __restrict__ cnt,
                       const float* __restrict__ sw, const float* __restrict__ sb,
                       float* __restrict__ h, float* __restrict__ stats) {
    int b = blockIdx.y;
    int n = blockIdx.x * blockDim.x + threadIdx.x;
    float s = 0.0f, sq = 0.0f;
    if (n < NN) {
        float inv = 1.0f / fmaxf(cnt[n], 1.0f);
        float a[CIN];
        #pragma unroll
        for (int c = 0; c < CIN; c++) a[c] = summed[(size_t)n * 512 + b * CIN + c] * inv;
        #pragma unroll
        for (int m = 0; m < MIDC; m++) {
            float acc = sb[m];
            #pragma unroll
            for (int c = 0; c < CIN; c++) acc += a[c] * sw[c * MIDC + m];
            acc = fmaxf(acc, 0.0f);
            h[(size_t)n * (BB * MIDC) + b * MIDC + m] = acc;
            s += acc; sq += acc * acc;
        }
    }
    __shared__ float rs[256], rq[256];
    int t = threadIdx.x;
    rs[t] = s; rq[t] = sq;
    __syncthreads();
    for (int off = 128; off > 0; off >>= 1) {
        if (t < off) { rs[t] += rs[t + off]; rq[t] += rq[t + off]; }
        __syncthreads();
    }
    if (t == 0) { atomicAdd(&stats[b], rs[0]); atomicAdd(&stats[BB + b], rq[0]); }
}

// ---------------- LayerNorm -> conv1+relu+BN -> conv2+relu+BN -> flatten into Z ----------------
__global__ void k_fused(const float* __restrict__ h, const float* __restrict__ stats,
                        const float* __restrict__ lng, const float* __restrict__ lnb,
                        const float* __restrict__ c1w, const float* __restrict__ c1b,
                        const float* __restrict__ bn1g, const float* __restrict__ bn1b,
                        const float* __restrict__ bn1m, const float* __restrict__ bn1v,
                        const float* __restrict__ c2w, const float* __restrict__ c2b,
                        const float* __restrict__ bn2g, const float* __restrict__ bn2b,
                        const float* __restrict__ bn2m, const float* __restrict__ bn2v,
                        float* __restrict__ Z) {
    int b = blockIdx.y;
    int n = blockIdx.x * blockDim.x + threadIdx.x;
    if (n >= NN) return;
    const float cf = (float)(NN * MIDC);
    float mu  = stats[b] / cf;
    float var = stats[BB + b] / cf - mu * mu;
    float rstd = rsqrtf(var + EPSV);
    float v[MIDC];
    #pragma unroll
    for (int m = 0; m < MIDC; m++) {
        float hv = h[(size_t)n * (BB * MIDC) + b * MIDC + m];
        v[m] = (hv - mu) * rstd * lng[n * MIDC + m] + lnb[n * MIDC + m];
    }
    float t1[G1C];
    #pragma unroll
    for (int o = 0; o < G1C; o++) {
        float acc = c1b[o];
        #pragma unroll
        for (int m = 0; m < MIDC; m++) acc += c1w[o * MIDC + m] * v[m];
        acc = fmaxf(acc, 0.0f);                                 // relu then BN
        t1[o] = (acc - bn1m[o]) * (bn1g[o] * rsqrtf(bn1v[o] + EPSV)) + bn1b[o];
    }
    #pragma unroll
    for (int g = 0; g < G2C; g++) {
        float acc = c2b[g];
        #pragma unroll
        for (int o = 0; o < G1C; o++) acc += c2w[g * G1C + o] * t1[o];
        acc = fmaxf(acc, 0.0f);                                 // relu then BN
        float z = (acc - bn2m[g]) * (bn2g[g] * rsqrtf(bn2v[g] + EPSV)) + bn2b[g];
        Z[(size_t)b * D0 + g * NN + n] = z;
    }
}

// ---------------- f32 WMMA GEMM partial: Oraw += A[64,K] * W[Nout,K]^T (K-split) ----------------
// Block = 128 threads = 4 waves; wave w owns row-tile w (M=64). All 4 waves share one
// 16-col weight tile, cooperatively staged into double-buffered LDS (async-to-LDS when
// available) so the dominant weight stream is read from L2/HBM exactly once.
// grid = (Nout/16, KSPLIT); Kpart = K / KSPLIT must be a multiple of KC.
__global__ void __launch_bounds__(128)
k_gemm_partial(const float* __restrict__ A, const float* __restrict__ W,
               float* __restrict__ Oraw, int Nout, int K, int Kpart) {
    __shared__ __align__(16) float lb[2][16 * LDSTRIDE];
    const int t    = threadIdx.x;
    const int l    = t & 31;          // lane
    const int wv   = t >> 5;          // wave id == row tile
    const int ct   = blockIdx.x;      // col tile
    const int kbase = blockIdx.y * Kpart;
    const int mn   = l & 15;
    const int koff = (l >> 4) << 1;   // 0 or 2 (ISA 7.12.2 f32 A/B fragment layout)
    const float* arow = A + (size_t)(wv * 16 + mn) * K + kbase + koff;
    const float* wbase = W + (size_t)(ct * 16) * K + kbase;

    // cooperative loader: 16 rows x KC floats = 256 x b128 segments, 2 per thread
    auto loadB = [&](int buf, int kc0) {
        #pragma unroll
        for (int s = t; s < 256; s += 128) {
            int r = s >> 4;
            int c = (s & 15) << 2;    // float offset within row
            const float* g = wbase + (size_t)r * K + kc0 + c;
            float* d = &lb[buf][r * LDSTRIDE + c];
#ifdef HAS_ASYNC_LDS
            __builtin_amdgcn_global_load_async_to_lds_b128(
                (AS1 b128_t*)g, (AS3 b128_t*)d, 0, 0);
#else
            *(float4*)d = *(const float4*)g;
#endif
        }
    };

    v8f acc0 = {0.f,0.f,0.f,0.f,0.f,0.f,0.f,0.f};
    v8f acc1 = {0.f,0.f,0.f,0.f,0.f,0.f,0.f,0.f};

    const int nchunks = Kpart / KC;
    loadB(0, 0);
    for (int c = 0; c < nchunks; ++c) {
        const int buf = c & 1;
#ifdef HAS_ASYNC_LDS
#if __has_builtin(__builtin_amdgcn_s_wait_asynccnt)
        __builtin_amdgcn_s_wait_asynccnt(0);
#else
        asm volatile("s_wait_asynccnt 0x0" ::: "memory");
#endif
#endif
        __syncthreads();                      // chunk c staged & visible to all 4 waves
        if (c + 1 < nchunks) loadB(buf ^ 1, (c + 1) * KC);  // prefetch next chunk
        const float* ar = arow + c * KC;
        const float* br = &lb[buf][mn * LDSTRIDE + koff];
        #pragma unroll
        for (int k = 0; k < KC; k += 8) {
            v2f a0 = *(const v2f*)(ar + k);
            v2f b0 = *(const v2f*)(br + k);
            v2f a1 = *(const v2f*)(ar + k + 4);
            v2f b1 = *(const v2f*)(br + k + 4);
            acc0 = __builtin_amdgcn_wmma_f32_16x16x4_f32(false, a0, false, b0,
                                                         (short)0, acc0, false, false);
            acc1 = __builtin_amdgcn_wmma_f32_16x16x4_f32(false, a1, false, b1,
                                                         (short)0, acc1, false, false);
        }
        __syncthreads();                      // done reading buf before it is re-staged
    }

    const int j = ct * 16 + mn;
    const int rbase = wv * 16 + ((l >> 4) << 3);   // rows v + 8*(l>>4)
    #pragma unroll
    for (int vi = 0; vi < 8; vi++) {
        atomicAdd(&Oraw[(size_t)(rbase + vi) * Nout + j], acc0[vi] + acc1[vi]);
    }
}

// ---------------- epilogue: out = relu(BN(raw + bias)) ----------------
__global__ void k_bias_bn_relu(const float* __restrict__ raw, const float* __restrict__ bias,
                               const float* __restrict__ g,  const float* __restrict__ be,
                               const float* __restrict__ mm, const float* __restrict__ vv,
                               float* __restrict__ out, int total, int Nout) {
    int i = blockIdx.x * blockDim.x + threadIdx.x;
    if (i >= total) return;
    int j = i % Nout;
    float scale = g[j] * rsqrtf(vv[j] + EPSV);
    float shift = be[j] - mm[j] * scale;
    float val = raw[i] + bias[j];
    out[i] = fmaxf(val * scale + shift, 0.0f);    // BN then relu (fc ordering)
}

// ---------------- final linear + softmax, writes (softmax, logits) ----------------
__global__ void k_head(const float* __restrict__ z2, const float* __restrict__ W,
                       const float* __restrict__ bias, float* __restrict__ out) {
    int b = threadIdx.x;
    if (b >= BB) return;
    float logits[OUTC];
    float mx = -1e30f;
    #pragma unroll
    for (int j = 0; j < OUTC; j++) {
        float acc = bias[j];
        for (int k = 0; k < FC2C; k++) acc += z2[b * FC2C + k] * W[j * FC2C + k];
        logits[j] = acc;
        mx = fmaxf(mx, acc);
    }
    float ex[OUTC], se = 0.0f;
    #pragma unroll
    for (int j = 0; j < OUTC; j++) { ex[j] = __expf(logits[j] - mx); se += ex[j]; }
    float inv = 1.0f / se;
    #pragma unroll
    for (int j = 0; j < OUTC; j++) {
        out[b * OUTC + j] = ex[j] * inv;                 // softmax first
        out[BB * OUTC + b * OUTC + j] = logits[j];       // then raw logits
    }
}

extern "C" void kernel_launch(void* const* d_in, const int* in_sizes, int n_in,
                              void* d_out, int out_size, void* d_ws, size_t ws_size,
                              hipStream_t stream) {
    const float* x    = (const float*)d_in[0];
    const int*   ei   = (const int*)  d_in[1];
    const float* ewr  = (const float*)d_in[2];
    // d_in[3] = sage_self_w (alpha 0 in reference; unused)
    const float* sw   = (const float*)d_in[4];
    const float* sb   = (const float*)d_in[5];
    const float* lng  = (const float*)d_in[6];
    const float* lnb  = (const float*)d_in[7];
    const float* c1w  = (const float*)d_in[8];
    const float* c1b  = (const float*)d_in[9];
    const float* bn1g = (const float*)d_in[10];
    const float* bn1b = (const float*)d_in[11];
    const float* bn1m = (const float*)d_in[12];
    const float* bn1v = (const float*)d_in[13];
    const float* c2w  = (const float*)d_in[14];
    const float* c2b  = (const float*)d_in[15];
    const float* bn2g = (const float*)d_in[16];
    const float* bn2b = (const float*)d_in[17];
    const float* bn2m = (const float*)d_in[18];
    const float* bn2v = (const float*)d_in[19];
    const float* fw1  = (const float*)d_in[20];
    const float* fb1  = (const float*)d_in[21];
    const float* bf1g = (const float*)d_in[22];
    const float* bf1b = (const float*)d_in[23];
    const float* bf1m = (const float*)d_in[24];
    const float* bf1v = (const float*)d_in[25];
    const float* fw2  = (const float*)d_in[26];
    const float* fb2  = (const float*)d_in[27];
    const float* bf2g = (const float*)d_in[28];
    const float* bf2b = (const float*)d_in[29];
    const float* bf2m = (const float*)d_in[30];
    const float* bf2v = (const float*)d_in[31];
    const float* hw   = (const float*)d_in[32];
    const float* hb   = (const float*)d_in[33];

    // workspace layout (floats); summed..z2raw contiguous for one-shot zeroing
    float* ws     = (float*)d_ws;
    float* summed = ws;                             // N*B*CIN = 1,536,000
    float* cnt    = summed + (size_t)NN * BB * CIN; // 3,000
    float* stats  = cnt + NN;                       // 128
    float* z1raw  = stats + 2 * BB;                 // 32,768
    float* z2raw  = z1raw + BB * FC1C;              // 8,192
    float* h      = z2raw + BB * FC2C;              // 3,072,000
    float* Z      = h + (size_t)NN * BB * MIDC;     // 1,536,000
    float* z1a    = Z + (size_t)BB * D0;            // 32,768
    float* z2a    = z1a + BB * FC1C;                // 8,192

    // 1. zero all atomic accumulators in one contiguous sweep
    int zn = NN * BB * CIN + NN + 2 * BB + BB * FC1C + BB * FC2C;
    k_zero<<<(zn + 255) / 256, 256, 0, stream>>>(summed, zn);

    // 2. weighted message scatter-sum + degree counts
    long long sthreads = (long long)EE * 128;
    k_scatter<<<(int)((sthreads + 255) / 256), 256, 0, stream>>>(x, ei, ewr, summed, cnt);

    // 3. mean-agg -> sage matmul -> relu -> h ; LN stats
    dim3 gnb((NN + 255) / 256, BB);
    k_sage<<<gnb, 256, 0, stream>>>(summed, cnt, sw, sb, h, stats);

    // 4. LN -> conv1 -> conv2 -> flattened Z [B, 24000]
    k_fused<<<gnb, 256, 0, stream>>>(h, stats, lng, lnb,
                                     c1w, c1b, bn1g, bn1b, bn1m, bn1v,
                                     c2w, c2b, bn2g, bn2b, bn2m, bn2v, Z);

    // 5. fc1: [64,24000] x [24000,512]^T, f32 WMMA, K split 25 ways (3200 waves)
    //    Kpart = 24000/25 = 960 = 15 chunks of 64
    k_gemm_partial<<<dim3(FC1C / 16, 25), 128, 0, stream>>>(Z, fw1, z1raw, FC1C, D0, D0 / 25);
    k_bias_bn_relu<<<(BB * FC1C + 255) / 256, 256, 0, stream>>>(
        z1raw, fb1, bf1g, bf1b, bf1m, bf1v, z1a, BB * FC1C, FC1C);

    // 6. fc2: [64,512] x [512,128]^T, K split 2 ways (Kpart = 256 = 4 chunks)
    k_gemm_partial<<<dim3(FC2C / 16, 2), 128, 0, stream>>>(z1a, fw2, z2raw, FC2C, FC1C, FC1C / 2);
    k_bias_bn_relu<<<(BB * FC2C + 255) / 256, 256, 0, stream>>>(
        z2raw, fb2, bf2g, bf2b, bf2m, bf2v, z2a, BB * FC2C, FC2C);

    // 7. final linear + softmax -> d_out = [softmax(64x10), logits(64x10)]
    k_head<<<1, BB, 0, stream>>>(z2a, hw, hb, (float*)d_out);
}